// Critic_7971459301587
// MI455X (gfx1250) — compile-verified
//
#include <hip/hip_runtime.h>
#include <hip/hip_bf16.h>

typedef __attribute__((ext_vector_type(2))) float v2f;
typedef __attribute__((ext_vector_type(8))) float v8f;

#define N_NODES   50000
#define N_EDGES   800000
#define NUM_GRAPHS 64
#define D         128
#define SLOPE     0.01f

__device__ __forceinline__ float leaky(float v) { return v >= 0.0f ? v : SLOPE * v; }

// ---------------- utility: zero a float array ----------------
__global__ void zero_f(float* __restrict__ p, int n) {
    int i = blockIdx.x * blockDim.x + threadIdx.x;
    if (i < n) p[i] = 0.0f;
}

// ---------------- degree (col-side, unit edge weights) ----------------
__global__ void deg_kernel(const int* __restrict__ col, float* __restrict__ deg) {
    int e = blockIdx.x * blockDim.x + threadIdx.x;
    if (e < N_EDGES) unsafeAtomicAdd(&deg[col[e]], 1.0f);
}

__global__ void dinv_kernel(const float* __restrict__ deg, float* __restrict__ dinv) {
    int n = blockIdx.x * blockDim.x + threadIdx.x;
    if (n < N_NODES) dinv[n] = rsqrtf(deg[n] + 1.0f);   // +1 self loop
}

// ---------------- GEMM: Y[M,128] = X[M,128] @ W[128,128], optional leaky ----
// One wave computes one 16x16 output tile with V_WMMA_F32_16X16X4_F32.
// M = 50000 = 3125*16 exactly; N = K = 128. EXEC all-ones (no divergence).
__global__ __launch_bounds__(256) void gemm_kernel(const float* __restrict__ X,
                                                   const float* __restrict__ W,
                                                   float* __restrict__ Y,
                                                   int applyLeaky) {
    const int lane  = threadIdx.x & 31;
    const int wave  = threadIdx.x >> 5;
    const int gw    = blockIdx.x * 8 + wave;     // 25000 waves total
    const int tileM = gw >> 3;                   // 0..3124
    const int tileN = gw & 7;                    // 0..7

    const int mrow  = tileM * 16 + (lane & 15);  // A row for this lane
    const int ncol  = tileN * 16 + (lane & 15);  // B col for this lane
    const int khalf = (lane >> 4) * 2;           // lanes 16-31 hold K+2,K+3

    const float* __restrict__ xp = X + mrow * D + khalf;
    const float* __restrict__ wp = W + khalf * D + ncol;

    v8f c = {};
#pragma unroll
    for (int k0 = 0; k0 < D; k0 += 4) {
        v2f a, b;
        a.x = xp[k0];
        a.y = xp[k0 + 1];
        b.x = wp[k0 * D];           // W[(k0+khalf  )*D + ncol]
        b.y = wp[k0 * D + D];       // W[(k0+khalf+1)*D + ncol]
        // 8 args: (neg_a, A, neg_b, B, c_mod, C, reuse_a, reuse_b)
        c = __builtin_amdgcn_wmma_f32_16x16x4_f32(false, a, false, b,
                                                  (short)0, c, false, false);
    }

    // C/D layout: VGPR r -> M=r (lanes 0-15) / M=r+8 (lanes 16-31), N=lane&15
    const int mBase = tileM * 16 + ((lane >> 4) << 3);
#pragma unroll
    for (int r = 0; r < 8; ++r) {
        float v = c[r];
        if (applyLeaky) v = leaky(v);
        Y[(mBase + r) * D + ncol] = v;
    }
}

// ---------------- edge scatter: agg[col] += h[row] * dinv[row]*dinv[col] ----
// One wave per edge; each lane handles 4 features (float4) -> 128 f32 atomics.
// h and agg are both 25.6 MB -> resident in the 192 MB L2.
__global__ __launch_bounds__(256) void scatter_kernel(const float* __restrict__ h,
                                                      const int* __restrict__ row,
                                                      const int* __restrict__ col,
                                                      const float* __restrict__ dinv,
                                                      float* __restrict__ agg) {
    const int lane = threadIdx.x & 31;
    const int e    = blockIdx.x * 8 + (threadIdx.x >> 5);
    if (e >= N_EDGES) return;
    const int   r    = row[e];
    const int   ci   = col[e];
    const float coef = dinv[r] * dinv[ci];

    const float4 v = *(const float4*)(h + r * D + lane * 4);
    float* dst = agg + ci * D + lane * 4;
    unsafeAtomicAdd(dst + 0, v.x * coef);
    unsafeAtomicAdd(dst + 1, v.y * coef);
    unsafeAtomicAdd(dst + 2, v.z * coef);
    unsafeAtomicAdd(dst + 3, v.w * coef);
}

// ---------------- combine: out = leaky(agg + h * dinv^2) ----------------
__global__ void combine_kernel(const float* __restrict__ h,
                               const float* __restrict__ agg,
                               const float* __restrict__ dinv,
                               float* __restrict__ out) {
    int i = blockIdx.x * blockDim.x + threadIdx.x;
    if (i < N_NODES * D) {
        int n = i >> 7;
        float di = dinv[n];
        out[i] = leaky(agg[i] + h[i] * di * di);
    }
}

// ---------------- fc2 (128 -> 1) + per-graph pooling accumulation --------
// One wave per node; 4 features/lane; wave32 xor-shuffle reduction.
__global__ __launch_bounds__(256) void fc2_pool_kernel(const float* __restrict__ y1,
                                                       const float* __restrict__ wfc2,
                                                       const int* __restrict__ batch,
                                                       float* __restrict__ sums,
                                                       float* __restrict__ cnts) {
    const int lane = threadIdx.x & 31;
    const int n    = blockIdx.x * 8 + (threadIdx.x >> 5);
    if (n >= N_NODES) return;
    const float4 a = *(const float4*)(y1 + n * D + lane * 4);
    const float4 w = *(const float4*)(wfc2 + lane * 4);
    float p = a.x * w.x + a.y * w.y + a.z * w.z + a.w * w.w;
#pragma unroll
    for (int off = 16; off > 0; off >>= 1) p += __shfl_xor(p, off, 32);
    if (lane == 0) {
        float v = leaky(p);
        int g = batch[n];
        unsafeAtomicAdd(&sums[g], v);
        unsafeAtomicAdd(&cnts[g], 1.0f);
    }
}

__global__ void finalize_kernel(const float* __restrict__ sums,
                                const float* __restrict__ cnts,
                                float* __restrict__ out) {
    int g = threadIdx.x;
    if (g < NUM_GRAPHS) out[g] = sums[g] / fmaxf(cnts[g], 1.0f);
}

// =========================================================================
extern "C" void kernel_launch(void* const* d_in, const int* in_sizes, int n_in,
                              void* d_out, int out_size, void* d_ws, size_t ws_size,
                              hipStream_t stream) {
    const float* x     = (const float*)d_in[0];
    const int*   ei    = (const int*)d_in[1];
    const int*   row   = ei;              // edge_index[0]
    const int*   col   = ei + N_EDGES;    // edge_index[1]
    const int*   batch = (const int*)d_in[2];
    const float* W0    = (const float*)d_in[3];
    const float* W1    = (const float*)d_in[4];
    const float* W2    = (const float*)d_in[5];
    const float* Wfc1  = (const float*)d_in[6];
    const float* Wfc2  = (const float*)d_in[7];

    char* ws = (char*)d_ws;
    const size_t NM = (size_t)N_NODES * D * sizeof(float);   // 25,600,000 (256-aligned)
    float* bufH   = (float*)(ws);                 // h = x @ W
    float* bufAgg = (float*)(ws + NM);            // neighbor aggregation
    float* bufX   = (float*)(ws + 2 * NM);        // layer output / next input
    float* deg    = (float*)(ws + 3 * NM);
    float* dinv   = (float*)(ws + 3 * NM + 204800);
    float* sums   = (float*)(ws + 3 * NM + 409600);          // 64 floats
    float* cnts   = sums + 64;                               // 64 floats

    const int TPB = 256;
    const int ndBlocks = (N_NODES * D + TPB - 1) / TPB;

    // degrees (with self loop) -> dinv
    zero_f<<<(N_NODES + TPB - 1) / TPB, TPB, 0, stream>>>(deg, N_NODES);
    deg_kernel<<<(N_EDGES + TPB - 1) / TPB, TPB, 0, stream>>>(col, deg);
    dinv_kernel<<<(N_NODES + TPB - 1) / TPB, TPB, 0, stream>>>(deg, dinv);

    // 3 GCN layers
    const float* cur = x;
    const float* Ws[3] = {W0, W1, W2};
    for (int l = 0; l < 3; ++l) {
        gemm_kernel<<<3125, 256, 0, stream>>>(cur, Ws[l], bufH, 0);
        zero_f<<<ndBlocks, TPB, 0, stream>>>(bufAgg, N_NODES * D);
        scatter_kernel<<<N_EDGES / 8, 256, 0, stream>>>(bufH, row, col, dinv, bufAgg);
        combine_kernel<<<ndBlocks, TPB, 0, stream>>>(bufH, bufAgg, dinv, bufX);
        cur = bufX;
    }

    // fc1 with leaky fused into the GEMM epilogue
    gemm_kernel<<<3125, 256, 0, stream>>>(cur, Wfc1, bufH, 1);

    // fc2 + mean pool
    zero_f<<<1, 128, 0, stream>>>(sums, 128);   // sums + cnts are adjacent
    fc2_pool_kernel<<<N_NODES / 8, 256, 0, stream>>>(bufH, Wfc2, batch, sums, cnts);
    finalize_kernel<<<1, 64, 0, stream>>>(sums, cnts, (float*)d_out);
}